// PatternMatchingInference_38525856645223
// MI455X (gfx1250) — compile-verified
//
#include <hip/hip_runtime.h>
#include <hip/hip_bf16.h>
#include <math.h>

// Problem constants (from reference): SIZE=68, SSN=8 -> C=9, S1=69, BATCH=256
#define C_        9
#define S1_       69
#define NE_       621        // C_*S1_   (elements of one scan-state row)
#define NSTEP_    68
#define NROW_     69         // NSTEP_+1
#define QCN_      4761       // S1_*S1_       (Q stride over cn, P stride over cn)
#define QCP_      42849      // C_*S1_*S1_    (Q stride over cp)
#define PBS_      42849      // per-batch P stride = C_*S1_*S1_
#define ABS_      42849      // per-batch alpha stride = NROW_*NE_ (coincidentally equal)
#define BATCH_    256
#define GB_       2          // batches per scan workgroup (Q register reuse)
#define SCAN_T_   640        // 20 wave32 waves; covers 621 outputs

#define NEG_INF (-__builtin_huge_valf())

typedef __attribute__((address_space(1))) int  gint_t;   // global int
typedef __attribute__((address_space(3))) int  lint_t;   // LDS int

// ---------------- CDNA5 (gfx1250) async LDS<->global + split waitcnt helpers ----------------
// Builtin exists on this toolchain with typed (global int*, lds int*, imm, imm) params.
__device__ __forceinline__ void async_store_lds_to_global_b32(void* gaddr, void* laddr) {
#if defined(__gfx1250__) && __has_builtin(__builtin_amdgcn_global_store_async_from_lds_b32)
  __builtin_amdgcn_global_store_async_from_lds_b32(
      (gint_t*)gaddr, (lint_t*)laddr, 0, 0);
#elif defined(__gfx1250__)
  unsigned long long g = (unsigned long long)gaddr;
  unsigned l = (unsigned)(unsigned long long)(__attribute__((address_space(3))) char*)laddr;
  asm volatile("global_store_async_from_lds_b32 %0, %1, off"
               :: "v"(g), "v"(l) : "memory");
#else
  *(float*)gaddr = *(float*)laddr;
#endif
}

template <int N>
__device__ __forceinline__ void wait_asynccnt() {
#if defined(__gfx1250__) && __has_builtin(__builtin_amdgcn_s_wait_asynccnt)
  __builtin_amdgcn_s_wait_asynccnt(N);
#elif defined(__gfx1250__)
  asm volatile("s_wait_asynccnt %0" :: "i"(N) : "memory");
#endif
}

__device__ __forceinline__ void wait_dscnt0() {
#if defined(__gfx1250__) && __has_builtin(__builtin_amdgcn_s_wait_dscnt)
  __builtin_amdgcn_s_wait_dscnt(0);
#elif defined(__gfx1250__)
  asm volatile("s_wait_dscnt 0x0" ::: "memory");
#endif
}

// ---------------- Kernel 1: unnormalized max-plus scan ----------------
// u_k[b,cn,sn] = max_{cp,sp}( P[b,cn,sp,sn] + fma(T, Q[cp,cn,sp,sn], u_{k-1}[b,cp,sp]) )
// u rows are written directly into the alpha slots of d_out (row k = u_k), via async
// LDS->global stores double-buffered against the LDS state buffers.
__global__ void __launch_bounds__(SCAN_T_)
scan_kernel(const float* __restrict__ P, const float* __restrict__ Q,
            const float* __restrict__ pi, const float* __restrict__ Tp,
            float* __restrict__ U)
{
  __shared__ float2 la[2][SCAN_T_];   // [buffer][cp*69+sp] -> (batch b0, batch b0+1)

  const int t  = threadIdx.x;
  const int cn = t / S1_;
  const int sn = t - cn * S1_;
  const int b0 = blockIdx.x * GB_;
  const float T = Tp[0];

  float* u0 = U + (size_t)b0 * ABS_;  // batch b0 u/alpha block
  float* u1 = u0 + ABS_;              // batch b0+1

  // u_0: pi[cp] at sp==0, -inf elsewhere (identical for both batches)
  if (t < NE_) {
    float v = (sn == 0) ? pi[cn] : NEG_INF;
    la[0][t] = make_float2(v, v);
    wait_dscnt0();                              // own LDS store visible to async engine
    async_store_lds_to_global_b32(u0 + t, &la[0][t].x);
    async_store_lds_to_global_b32(u1 + t, &la[0][t].y);
  }
  __syncthreads();

  const float* qbase = Q + cn * QCN_ + sn;                  // + cp*QCP_ + sp*S1_
  const float* p0    = P + (size_t)b0 * PBS_ + cn * QCN_ + sn;  // + sp*S1_
  const float* p1    = p0 + PBS_;

  for (int k = 1; k <= NSTEP_; ++k) {
    const int cur = (k + 1) & 1;
    const int nxt = k & 1;
    wait_asynccnt<GB_>();   // async stores issued at step k-2 (from buffer `nxt`) are done
    if (t < NE_) {
      float best0 = NEG_INF, best1 = NEG_INF;
      const float2* lap = &la[cur][0];
      #pragma unroll 3
      for (int sp = 0; sp < S1_; ++sp) {
        float m0 = NEG_INF, m1 = NEG_INF;
        const float*  qp   = qbase + sp * S1_;
        const float2* lrow = lap + sp;
        #pragma unroll
        for (int cp = 0; cp < C_; ++cp) {
          float  q = qp[cp * QCP_];     // coalesced along sn; hot in 192MB L2
          float2 l = lrow[cp * S1_];    // LDS broadcast (same addr across lanes)
          m0 = fmaxf(m0, fmaf(T, q, l.x));
          m1 = fmaxf(m1, fmaf(T, q, l.y));
        }
        best0 = fmaxf(best0, p0[sp * S1_] + m0);
        best1 = fmaxf(best1, p1[sp * S1_] + m1);
      }
      la[nxt][t] = make_float2(best0, best1);
      wait_dscnt0();                    // own LDS write complete before async engine reads it
      async_store_lds_to_global_b32(u0 + k * NE_ + t, &la[nxt][t].x);
      async_store_lds_to_global_b32(u1 + k * NE_ + t, &la[nxt][t].y);
    }
    __syncthreads();
  }
  wait_asynccnt<0>();
}

// ---------------- Kernel 2: n_k = logsumexp(u_k); alpha[i] = u[68-i] + n[i] - n[68-i] (in place)
__global__ void __launch_bounds__(256)
finalize_kernel(float* __restrict__ U)
{
  __shared__ float nv[NROW_];
  const int t = threadIdx.x, lane = t & 31, wid = t >> 5;
  float* ub = U + (size_t)blockIdx.x * ABS_;

  for (int k = wid; k < NROW_; k += 8) {   // one wave32 per row
    const float* row = ub + k * NE_;
    float m = NEG_INF;
    for (int f = lane; f < NE_; f += 32) m = fmaxf(m, row[f]);
    #pragma unroll
    for (int off = 16; off > 0; off >>= 1) m = fmaxf(m, __shfl_xor(m, off, 32));
    float s = 0.f;
    for (int f = lane; f < NE_; f += 32) s += expf(row[f] - m);   // exp(-inf)=0 ok
    #pragma unroll
    for (int off = 16; off > 0; off >>= 1) s += __shfl_xor(s, off, 32);
    if (lane == 0) nv[k] = m + logf(s);
  }
  __syncthreads();

  for (int k = 0; k <= NSTEP_ / 2; ++k) {  // pairwise reversal, k=34 self-paired (dn=0)
    const int   k2 = NSTEP_ - k;
    const float dn = nv[k] - nv[k2];
    float* r0 = ub + k  * NE_;
    float* r1 = ub + k2 * NE_;
    for (int f = t; f < NE_; f += 256) {
      float a = r0[f], c = r1[f];
      r0[f] = c + dn;     // alpha[k]    = u[68-k] + n[k]    - n[68-k]
      r1[f] = a - dn;     // alpha[68-k] = u[k]    + n[68-k] - n[k]
    }
  }
}

// ---------------- Kernel 3: argmax backtrace (first-occurrence ties, like jnp.argmax)
__global__ void __launch_bounds__(256)
backtrace_kernel(const float* __restrict__ P, const float* __restrict__ Q,
                 const float* __restrict__ Tp, const int* __restrict__ ls,
                 const float* __restrict__ alpha, float* __restrict__ mp_out)
{
  __shared__ float rv[8];
  __shared__ int   ri[8];
  __shared__ int   c_arr[NROW_], l_arr[NROW_], t_arr[NROW_];
  __shared__ int   cur_ct[2];

  const int b = blockIdx.x, t = threadIdx.x;
  const int lane = t & 31, wid = t >> 5;
  const float T = Tp[0];

  if (t == 0) {
    c_arr[0] = 3; l_arr[0] = 0; t_arr[0] = ls[b];
    cur_ct[0] = 3; cur_ct[1] = ls[b];
  }
  __syncthreads();

  const float* ab = alpha + (size_t)b * ABS_;
  const float* Pb = P + (size_t)b * PBS_;

  for (int k = 1; k <= NSTEP_; ++k) {
    const int c = cur_ct[0], tt = cur_ct[1];
    const float* arow = ab + k * NE_;
    float bv = NEG_INF; int bi = 0x7fffffff;
    for (int f = t; f < NE_; f += 256) {
      const int c2 = f / S1_;
      const int s  = f - c2 * S1_;
      const float pterm = Pb[c * QCN_ + s * S1_ + tt];
      const float qterm = Q[c * QCP_ + c2 * QCN_ + s * S1_ + tt];
      const float sc = pterm + fmaf(T, qterm, arow[f]);
      if (sc > bv || (sc == bv && f < bi)) { bv = sc; bi = f; }
    }
    #pragma unroll
    for (int off = 16; off > 0; off >>= 1) {
      float ov = __shfl_xor(bv, off, 32);
      int   oi = __shfl_xor(bi, off, 32);
      if (ov > bv || (ov == bv && oi < bi)) { bv = ov; bi = oi; }
    }
    if (lane == 0) { rv[wid] = bv; ri[wid] = bi; }
    __syncthreads();
    if (t == 0) {
      float fv = rv[0]; int fi = ri[0];
      #pragma unroll
      for (int w = 1; w < 8; ++w)
        if (rv[w] > fv || (rv[w] == fv && ri[w] < fi)) { fv = rv[w]; fi = ri[w]; }
      const int cnn = fi / S1_;
      const int tn  = fi - cnn * S1_;
      c_arr[k] = cnn; t_arr[k] = tn; l_arr[k] = tn - tt;
      cur_ct[0] = cnn; cur_ct[1] = tn;
    }
    __syncthreads();
  }

  // mp[b,j] = (c_{68-j}, l_{67-j} (0 at j=68), t_{68-j}), emitted as floats
  for (int j = t; j < NROW_; j += 256) {
    float* o = mp_out + ((size_t)b * NROW_ + j) * 3;
    o[0] = (float)c_arr[NSTEP_ - j];
    o[1] = (j == NSTEP_) ? 0.f : (float)l_arr[NSTEP_ - 1 - j];
    o[2] = (float)t_arr[NSTEP_ - j];
  }
}

// ---------------- Launch ----------------
extern "C" void kernel_launch(void* const* d_in, const int* in_sizes, int n_in,
                              void* d_out, int out_size, void* d_ws, size_t ws_size,
                              hipStream_t stream) {
  (void)in_sizes; (void)n_in; (void)d_ws; (void)ws_size; (void)out_size;
  const float* P  = (const float*)d_in[0];   // (256,1,9,69,69)
  const float* Q  = (const float*)d_in[1];   // (1,9,9,69,69)
  const float* pi = (const float*)d_in[2];   // (9,)
  const float* T  = (const float*)d_in[3];   // (1,)
  const int*   ls = (const int*)d_in[4];     // (256,)

  float* out   = (float*)d_out;
  float* alpha = out;                                   // 256*69*9*69 floats
  float* mp    = out + (size_t)BATCH_ * NROW_ * NE_;    // 256*69*3 values (as floats)

  scan_kernel     <<<BATCH_ / GB_, SCAN_T_, 0, stream>>>(P, Q, pi, T, alpha);
  finalize_kernel <<<BATCH_,       256,     0, stream>>>(alpha);
  backtrace_kernel<<<BATCH_,       256,     0, stream>>>(P, Q, T, ls, alpha, mp);
}